// Quantizer_81552839016742
// MI455X (gfx1250) — compile-verified
//
#include <hip/hip_runtime.h>
#include <hip/hip_bf16.h>

typedef __attribute__((ext_vector_type(2))) float v2f;
typedef __attribute__((ext_vector_type(8))) float v8f;

#define D_DIM 64
#define TILE_K 256                      // codewords staged in LDS per slab
#define PAIR_STRIDE (2 * TILE_K + 32)   // 544 words; 544%64==32 -> halves hit disjoint banks
#define ROWS_PER_WAVE 32                // two 16x16 M tiles
#define WAVES_PER_BLOCK 8
#define ROWS_PER_BLOCK (ROWS_PER_WAVE * WAVES_PER_BLOCK) // 256

// ---------------------------------------------------------------- utilities
__global__ void vq_zero_counts(int* counts, int K) {
    for (int k = threadIdx.x; k < K; k += blockDim.x) counts[k] = 0;
}

// ||e_k||^2 per codeword. e is [D,K] row-major -> coalesced across k.
__global__ void vq_enorm(const float* __restrict__ e, float* __restrict__ enorm, int K) {
    int k = blockIdx.x * blockDim.x + threadIdx.x;
    if (k >= K) return;
    float s = 0.f;
    #pragma unroll
    for (int d = 0; d < D_DIM; ++d) {
        float v = e[d * K + k];
        s += v * v;
    }
    enorm[k] = s;
}

// ------------------------------------------------- WMMA argmin (main GEMM)
// score(n,k) = ||e_k||^2 - 2 * (x_n . e_k)   (||x||^2 dropped: constant per row)
// LDS holds e pair-interleaved: (d,k) -> lds_e[(d>>1)*PAIR_STRIDE + 2k + (d&1)]
// so one B fragment (e[d,k], e[d+1,k]) is a single contiguous ds_load_b64.
__global__ void __launch_bounds__(256)
vq_argmin(const float* __restrict__ x, const float* __restrict__ e,
          const float* __restrict__ enorm, int* __restrict__ enc_idx, int K) {
    __shared__ float lds_e[32 * PAIR_STRIDE];   // 32 pairs x 544 words = 69.6 KB

    const int tid  = threadIdx.x;
    const int wave = tid >> 5;
    const int lane = tid & 31;
    const int col  = lane & 15;     // N-position within tile / k-column
    const int half = lane >> 4;     // selects K-pair for A/B fragments

    const int row0 = (blockIdx.x * WAVES_PER_BLOCK + wave) * ROWS_PER_WAVE;

    // ---- load A fragments for both 16-row tiles (reused for all 2048 k's)
    v2f a0[16], a1[16];
    {
        const float* xr0 = x + (size_t)(row0 + col) * D_DIM;
        const float* xr1 = x + (size_t)(row0 + 16 + col) * D_DIM;
        #pragma unroll
        for (int s = 0; s < 16; ++s) {
            int d = 4 * s + 2 * half;
            a0[s] = *(const v2f*)(xr0 + d);   // 8B aligned (d even, row 256B aligned)
            a1[s] = *(const v2f*)(xr1 + d);
        }
    }

    float bestv0[8], bestv1[8];
    int   besti0[8], besti1[8];
    #pragma unroll
    for (int r = 0; r < 8; ++r) {
        bestv0[r] = 3.4e38f; bestv1[r] = 3.4e38f;
        besti0[r] = 0;       besti1[r] = 0;
    }

    // fragment loader: one contiguous ds_load_b64 per (d,d+1) pair
    auto load_frags = [&](v2f* b, int koff) {
        #pragma unroll
        for (int s = 0; s < 16; ++s)
            b[s] = *(const v2f*)&lds_e[(2 * s + half) * PAIR_STRIDE + 2 * (koff + col)];
    };
    // 32-WMMA chain for one 16-codeword tile + running argmin update
    auto compute_tile = [&](const v2f* bf, int kglob_base) {
        v8f c0 = {}, c1 = {};
        #pragma unroll
        for (int s = 0; s < 16; ++s) {
            c0 = __builtin_amdgcn_wmma_f32_16x16x4_f32(
                    false, a0[s], false, bf[s], (short)0, c0, false, false);
            c1 = __builtin_amdgcn_wmma_f32_16x16x4_f32(
                    false, a1[s], false, bf[s], (short)0, c1, false, false);
        }
        const int kglob = kglob_base + col;
        const float en = enorm[kglob];
        #pragma unroll
        for (int r = 0; r < 8; ++r) {
            float s0 = en - 2.0f * c0[r];
            if (s0 < bestv0[r]) { bestv0[r] = s0; besti0[r] = kglob; }
            float s1 = en - 2.0f * c1[r];
            if (s1 < bestv1[r]) { bestv1[r] = s1; besti1[r] = kglob; }
        }
    };

    for (int slab = 0; slab < K; slab += TILE_K) {
        // ---- cooperative stage of e[0:64, slab:slab+256] into LDS (pair-interleaved)
        __syncthreads();
        {
            float4 ta[8], tb[8];
            #pragma unroll
            for (int i = 0; i < 8; ++i) {          // all 16 global loads in flight
                int li = tid + i * 256;            // 0..2047
                int p  = li >> 6;                  // d-pair 0..31
                int c4 = li & 63;                  // k-chunk of 4
                const float* g0 = e + (size_t)(2 * p) * K + slab + 4 * c4;
                ta[i] = *(const float4*)g0;        // row 2p
                tb[i] = *(const float4*)(g0 + K);  // row 2p+1
            }
            #pragma unroll
            for (int i = 0; i < 8; ++i) {          // zip rows into pair-interleaved LDS
                int li = tid + i * 256;
                int p  = li >> 6;
                int c4 = li & 63;
                float4 o0 = make_float4(ta[i].x, tb[i].x, ta[i].y, tb[i].y);
                float4 o1 = make_float4(ta[i].z, tb[i].z, ta[i].w, tb[i].w);
                float* base = &lds_e[p * PAIR_STRIDE + 8 * c4];
                *(float4*)(base)     = o0;
                *(float4*)(base + 4) = o1;
            }
        }
        __syncthreads();

        // ---- ping-pong double buffer: no register rotation needed
        v2f bufA[16], bufB[16];
        load_frags(bufA, 0);
        for (int koff = 0; koff < TILE_K; koff += 32) {
            load_frags(bufB, koff + 16);           // prefetch odd tile
            compute_tile(bufA, slab + koff);       // compute even tile
            if (koff + 32 < TILE_K)
                load_frags(bufA, koff + 32);       // prefetch next even tile
            compute_tile(bufB, slab + koff + 16);  // compute odd tile
        }
    }

    // ---- cross-lane argmin within each 16-lane half (first-min tie-break)
    #pragma unroll
    for (int r = 0; r < 8; ++r) {
        float bv0 = bestv0[r]; int bi0 = besti0[r];
        float bv1 = bestv1[r]; int bi1 = besti1[r];
        #pragma unroll
        for (int m = 1; m <= 8; m <<= 1) {
            float ov0 = __shfl_xor(bv0, m, 32); int oi0 = __shfl_xor(bi0, m, 32);
            if (ov0 < bv0 || (ov0 == bv0 && oi0 < bi0)) { bv0 = ov0; bi0 = oi0; }
            float ov1 = __shfl_xor(bv1, m, 32); int oi1 = __shfl_xor(bi1, m, 32);
            if (ov1 < bv1 || (ov1 == bv1 && oi1 < bi1)) { bv1 = ov1; bi1 = oi1; }
        }
        if (col == 0) {
            enc_idx[row0 +      r + 8 * half] = bi0;  // tile 0, row M = r + 8*half
            enc_idx[row0 + 16 + r + 8 * half] = bi1;  // tile 1
        }
    }
}

// ------------------------------------- gather + per-block loss partial sums
// wave-per-row (8 rows per wave), deterministic fixed-order reductions.
__global__ void __launch_bounds__(256)
vq_gather(const float* __restrict__ x, const float* __restrict__ e,
          const int* __restrict__ enc_idx, float* __restrict__ out,
          int* __restrict__ counts, float* __restrict__ partials, int K) {
    __shared__ float wsum[WAVES_PER_BLOCK];
    const int wave = threadIdx.x >> 5;
    const int lane = threadIdx.x & 31;
    const int rowBase = blockIdx.x * 64 + wave * 8;

    float acc = 0.f;
    #pragma unroll
    for (int i = 0; i < 8; ++i) {
        int n = rowBase + i;
        int idx = enc_idx[n];
        float q0 = e[(size_t)lane * K + idx];
        float q1 = e[(size_t)(lane + 32) * K + idx];
        float xv0 = x[(size_t)n * D_DIM + lane];
        float xv1 = x[(size_t)n * D_DIM + lane + 32];
        out[(size_t)n * D_DIM + lane]      = q0;   // quantized_st == quantized (fwd)
        out[(size_t)n * D_DIM + lane + 32] = q1;
        float d0 = q0 - xv0, d1 = q1 - xv1;
        acc += d0 * d0 + d1 * d1;
        if (lane == 0) atomicAdd(&counts[idx], 1); // integer: exact, deterministic
    }
    #pragma unroll
    for (int m = 16; m >= 1; m >>= 1) acc += __shfl_xor(acc, m, 32);
    if (lane == 0) wsum[wave] = acc;
    __syncthreads();
    if (threadIdx.x == 0) {
        float s = 0.f;
        #pragma unroll
        for (int w = 0; w < WAVES_PER_BLOCK; ++w) s += wsum[w];  // fixed order
        partials[blockIdx.x] = s;
    }
}

// -------------------------------------------- final loss + perplexity (1 CTA)
__global__ void __launch_bounds__(256)
vq_finalize(const float* __restrict__ partials, int nPart,
            const int* __restrict__ counts, int K,
            float* __restrict__ out, size_t nd, float invN, float invND) {
    __shared__ float red[256];
    const int tid = threadIdx.x;

    float s = 0.f;
    for (int i = tid; i < nPart; i += 256) s += partials[i];   // fixed order per thread
    red[tid] = s; __syncthreads();
    #pragma unroll
    for (int off = 128; off >= 1; off >>= 1) {
        if (tid < off) red[tid] += red[tid + off];
        __syncthreads();
    }
    float loss = 1.25f * red[0] * invND;   // q_loss + 0.25*e_loss, both == mean((q-x)^2)
    __syncthreads();

    float h = 0.f;
    for (int k = tid; k < K; k += 256) {
        float p = (float)counts[k] * invN;
        h += p * logf(p + 1e-10f);
    }
    red[tid] = h; __syncthreads();
    #pragma unroll
    for (int off = 128; off >= 1; off >>= 1) {
        if (tid < off) red[tid] += red[tid + off];
        __syncthreads();
    }
    if (tid == 0) {
        out[nd]     = loss;
        out[nd + 1] = expf(-red[0]);
    }
}

// ---------------------------------------------------------------- launcher
extern "C" void kernel_launch(void* const* d_in, const int* in_sizes, int n_in,
                              void* d_out, int out_size, void* d_ws, size_t ws_size,
                              hipStream_t stream) {
    const float* x = (const float*)d_in[0];   // [N, 64]
    const float* e = (const float*)d_in[1];   // [64, K]
    float* out = (float*)d_out;

    const int N = in_sizes[0] / D_DIM;        // 131072
    const int K = in_sizes[1] / D_DIM;        // 2048
    const int nGatherBlocks = N / 64;

    // workspace carve-up (all 4B aligned; ~0.55 MB total)
    float* enorm    = (float*)d_ws;                  // K floats
    int*   counts   = (int*)(enorm + K);             // K ints
    float* partials = (float*)(counts + K);          // nGatherBlocks floats
    int*   enc_idx  = (int*)(partials + nGatherBlocks); // N ints
    (void)ws_size; (void)n_in; (void)out_size;

    vq_zero_counts<<<1, 256, 0, stream>>>(counts, K);
    vq_enorm<<<(K + 255) / 256, 256, 0, stream>>>(e, enorm, K);
    vq_argmin<<<N / ROWS_PER_BLOCK, 256, 0, stream>>>(x, e, enorm, enc_idx, K);
    vq_gather<<<nGatherBlocks, 256, 0, stream>>>(x, e, enc_idx, out, counts, partials, K);
    vq_finalize<<<1, 256, 0, stream>>>(partials, nGatherBlocks, counts, K, out,
                                       (size_t)N * D_DIM, 1.0f / (float)N,
                                       1.0f / ((float)N * (float)D_DIM));
}